// AutoEncoder_29231547417125
// MI455X (gfx1250) — compile-verified
//
#include <hip/hip_runtime.h>

#define XS   128
#define XS2  (128 * 128)
#define XS3  (128 * 128 * 128)
#define BATCH 8
#define NPTS  262144

typedef float v2f __attribute__((ext_vector_type(2)));
typedef float v8f __attribute__((ext_vector_type(8)));

// ---------------- zero workspace (grid + accumulators) ----------------
__global__ void zero_kernel(float4* __restrict__ ws, long n4) {
    long t = (long)blockIdx.x * blockDim.x + threadIdx.x;
    if (t < n4) ws[t] = make_float4(0.f, 0.f, 0.f, 0.f);
}

// ---------------- scatter-add points into per-batch grids ----------------
__global__ __launch_bounds__(256) void scatter_kernel(const int* __restrict__ idx,
                                                      const float* __restrict__ val,
                                                      float* __restrict__ grid) {
    int t = blockIdx.x * blockDim.x + threadIdx.x;   // 0 .. B*N-1 (exact multiple of 256)
    int b = t >> 18;                                 // N = 2^18
    const int* p = idx + (long)t * 3;
    int ix = p[0], iy = p[1], iz = p[2];
    float v = val[t];
    long off = (long)b * XS3 + (long)ix * XS2 + iy * XS + iz;
    unsafeAtomicAdd(grid + off, v);                  // global_atomic_add_f32, no return
}

// ---------------- diff reduction: tv = sum|d|, mse = sum d^2 ----------------
__global__ __launch_bounds__(256) void reduce_kernel(const float* __restrict__ grid,
                                                     float* __restrict__ acc) {
    const int b = blockIdx.y;
    const int r = blockIdx.x * 256 + threadIdx.x;    // 0 .. 16383 = (x,y) row id
    const int x = r >> 7;
    const int y = r & 127;

    const float* base = grid + (long)b * XS3 + (long)x * XS2 + (long)y * XS;
    const float4* cur = (const float4*)base;         // z-row, 512B aligned
    const float4* yn  = (const float4*)(base + XS);  // (x, y+1) row
    const float4* xn  = (const float4*)(base + XS2); // (x+1, y) row
    const bool hy = (y < XS - 1);
    const bool hx = (x < XS - 1);

    float tv = 0.f, ms = 0.f;
    float prev = 0.f;
    for (int zc = 0; zc < XS / 4; ++zc) {
        float4 c = cur[zc];
        float d;
        // d3: along z (127 diffs per row)
        if (zc) { d = c.x - prev; tv += fabsf(d); ms += d * d; }
        d = c.y - c.x; tv += fabsf(d); ms += d * d;
        d = c.z - c.y; tv += fabsf(d); ms += d * d;
        d = c.w - c.z; tv += fabsf(d); ms += d * d;
        prev = c.w;
        // d2: along y
        if (hy) {
            float4 n = yn[zc];
            d = n.x - c.x; tv += fabsf(d); ms += d * d;
            d = n.y - c.y; tv += fabsf(d); ms += d * d;
            d = n.z - c.z; tv += fabsf(d); ms += d * d;
            d = n.w - c.w; tv += fabsf(d); ms += d * d;
        }
        // d1: along x
        if (hx) {
            float4 n = xn[zc];
            d = n.x - c.x; tv += fabsf(d); ms += d * d;
            d = n.y - c.y; tv += fabsf(d); ms += d * d;
            d = n.z - c.z; tv += fabsf(d); ms += d * d;
            d = n.w - c.w; tv += fabsf(d); ms += d * d;
        }
    }

    // ---- wave32 cross-lane reduction via V_WMMA_F32_16X16X4_F32 ----
    // A (16x4 f32, 2 VGPRs/lane) = {partial, 0}; B (4x16) = all ones
    // (all-ones B is layout-independent). D[m][n] = partial(lane m) + partial(lane m+16).
    v2f bo; bo.x = 1.f; bo.y = 1.f;
    v8f c0 = {};
    v2f a;  a.x = tv; a.y = 0.f;
    v8f dtv = __builtin_amdgcn_wmma_f32_16x16x4_f32(false, a, false, bo,
                                                    (short)0, c0, false, false);
    a.x = ms;
    v8f dms = __builtin_amdgcn_wmma_f32_16x16x4_f32(false, a, false, bo,
                                                    (short)0, c0, false, false);
    // lanes 0-15 hold D rows 0..7, lanes 16-31 hold rows 8..15
    float stv = dtv[0] + dtv[1] + dtv[2] + dtv[3] + dtv[4] + dtv[5] + dtv[6] + dtv[7];
    float sms = dms[0] + dms[1] + dms[2] + dms[3] + dms[4] + dms[5] + dms[6] + dms[7];
    stv += __shfl_xor(stv, 16, 32);
    sms += __shfl_xor(sms, 16, 32);

    // ---- block accumulate in LDS (ds_add_f32), then 2 global atomics ----
    __shared__ float s_acc[2];
    if (threadIdx.x < 2) s_acc[threadIdx.x] = 0.f;
    __syncthreads();
    if ((threadIdx.x & 31) == 0) {
        atomicAdd(&s_acc[0], stv);
        atomicAdd(&s_acc[1], sms);
    }
    __syncthreads();
    if (threadIdx.x == 0) {
        unsafeAtomicAdd(&acc[b], s_acc[0]);        // tv raw sum
        unsafeAtomicAdd(&acc[8 + b], s_acc[1]);    // mse raw sum
    }
}

// ---------------- scale raw sums into outputs ----------------
__global__ void finalize_kernel(const float* __restrict__ acc, float* __restrict__ out) {
    int i = threadIdx.x;
    if (i < 16) {
        const float s = (i < 8) ? (1.f / 2097152.f)            // / xsize^3
                                : (1.f / 32512.f);             // / (2*128^2 - 2*128)
        out[i] = acc[i] * s;
    }
}

extern "C" void kernel_launch(void* const* d_in, const int* in_sizes, int n_in,
                              void* d_out, int out_size, void* d_ws, size_t ws_size,
                              hipStream_t stream) {
    const int*   idx = (const int*)d_in[0];
    const float* val = (const float*)d_in[1];
    // (d_in[2] = xsize scalar; fixed at 128 per reference)

    float* acc  = (float*)d_ws;      // 16 accumulators (+pad to 256B)
    float* grid = acc + 64;          // B * 128^3 floats = 64 MiB

    const long total_floats = (long)BATCH * XS3 + 64;
    const long n4 = total_floats / 4;
    zero_kernel<<<(unsigned)((n4 + 255) / 256), 256, 0, stream>>>((float4*)d_ws, n4);

    scatter_kernel<<<(BATCH * NPTS) / 256, 256, 0, stream>>>(idx, val, grid);

    dim3 rgrid(XS2 / 256, BATCH);    // 64 x 8 blocks, one (x,y) z-row per thread
    reduce_kernel<<<rgrid, 256, 0, stream>>>(grid, acc);

    finalize_kernel<<<1, 32, 0, stream>>>(acc, (float*)d_out);
}